// Decoder_G_12120397709392
// MI455X (gfx1250) — compile-verified
//
#include <hip/hip_runtime.h>

// ---------------------------------------------------------------------------
// Decoder: x = MLP(emb1 - knn_interp(emb2, h_pos2 -> h_pos1))
//          res = knn_interp(l_y1 - knn_interp(l_y2, l_pos2->l_pos1), l_pos1->h_pos1)
//          out = x + res        (all fp32, K=3 neighbors)
// Sizes: Nh=16384, Nl=4096, H=256, O=3.
// MLP GEMMs run on V_WMMA_F32_16X16X4_F32 (fp32 matrix pipe, wave32).
// Weights are pre-transposed once so both A and B fragments load as b64.
// ---------------------------------------------------------------------------

typedef __attribute__((ext_vector_type(2))) float v2f;
typedef __attribute__((ext_vector_type(8))) float v8f;

#define NH 16384
#define NL 4096
#define HDIM 256
#define ODIM 3
#define KNN_EPS 1e-16f

// ---------------- KNN top-3 (brute force, LDS-tiled) -----------------------
// One thread per query. Nq and Ns are multiples of 256 for every call here.
__global__ __launch_bounds__(256)
void knn3_kernel(const float* __restrict__ pos_y, const float* __restrict__ pos_x,
                 int Ns, int* __restrict__ out_idx, float* __restrict__ out_w)
{
    __shared__ float sx[256 * 3];
    const int q  = blockIdx.x * 256 + threadIdx.x;
    const float qx = pos_y[q * 3 + 0];
    const float qy = pos_y[q * 3 + 1];
    const float qz = pos_y[q * 3 + 2];

    float d0 = 3.4e38f, d1 = 3.4e38f, d2 = 3.4e38f;
    int   i0 = 0, i1 = 0, i2 = 0;

    for (int base = 0; base < Ns; base += 256) {
        __syncthreads();
#pragma unroll
        for (int t = 0; t < 3; ++t)
            sx[t * 256 + threadIdx.x] = pos_x[base * 3 + t * 256 + threadIdx.x];
        __syncthreads();
#pragma unroll 4
        for (int j = 0; j < 256; ++j) {
            const float dx = qx - sx[j * 3 + 0];
            const float dy = qy - sx[j * 3 + 1];
            const float dz = qz - sx[j * 3 + 2];
            const float d  = fmaf(dx, dx, fmaf(dy, dy, dz * dz));
            const int  jj  = base + j;
            if (d < d2) {
                if (d < d1) {
                    d2 = d1; i2 = i1;
                    if (d < d0) { d1 = d0; i1 = i0; d0 = d; i0 = jj; }
                    else        { d1 = d;  i1 = jj; }
                } else { d2 = d; i2 = jj; }
            }
        }
    }
    float w0 = 1.0f / fmaxf(d0, KNN_EPS);
    float w1 = 1.0f / fmaxf(d1, KNN_EPS);
    float w2 = 1.0f / fmaxf(d2, KNN_EPS);
    const float inv = 1.0f / (w0 + w1 + w2);
    out_idx[q * 3 + 0] = i0; out_idx[q * 3 + 1] = i1; out_idx[q * 3 + 2] = i2;
    out_w[q * 3 + 0] = w0 * inv; out_w[q * 3 + 1] = w1 * inv; out_w[q * 3 + 2] = w2 * inv;
}

// ------------- xin = emb1 - sum_k w_k * emb2[idx_k]  (H=256 features) ------
__global__ __launch_bounds__(256)
void interp_sub_emb_kernel(const float* __restrict__ emb1, const float* __restrict__ emb2,
                           const int* __restrict__ idx, const float* __restrict__ w,
                           float* __restrict__ out)
{
    const int q = blockIdx.x;
    const int f = threadIdx.x;
    const int i0 = idx[q * 3 + 0], i1 = idx[q * 3 + 1], i2 = idx[q * 3 + 2];
    const float w0 = w[q * 3 + 0], w1 = w[q * 3 + 1], w2 = w[q * 3 + 2];
    float v = w0 * emb2[i0 * HDIM + f];
    v = fmaf(w1, emb2[i1 * HDIM + f], v);
    v = fmaf(w2, emb2[i2 * HDIM + f], v);
    out[q * HDIM + f] = emb1[q * HDIM + f] - v;
}

// ------------- t = l_y1 - sum_k w_k * l_y2[idx_k]  (3 features) ------------
__global__ __launch_bounds__(256)
void interp_sub3_kernel(const float* __restrict__ y1, const float* __restrict__ y2,
                        const int* __restrict__ idx, const float* __restrict__ w,
                        float* __restrict__ out, int Nq)
{
    const int q = blockIdx.x * 256 + threadIdx.x;
    if (q >= Nq) return;
    const int i0 = idx[q * 3 + 0], i1 = idx[q * 3 + 1], i2 = idx[q * 3 + 2];
    const float w0 = w[q * 3 + 0], w1 = w[q * 3 + 1], w2 = w[q * 3 + 2];
#pragma unroll
    for (int c = 0; c < 3; ++c) {
        float v = w0 * y2[i0 * 3 + c];
        v = fmaf(w1, y2[i1 * 3 + c], v);
        v = fmaf(w2, y2[i2 * 3 + c], v);
        out[q * 3 + c] = y1[q * 3 + c] - v;
    }
}

// ------------- 256x256 transpose (one-time, LDS-tiled) ---------------------
// Bt[n][k] = B[k][n] so WMMA B-fragments become contiguous-K b64 loads.
__global__ __launch_bounds__(256)
void transpose_w_kernel(const float* __restrict__ B, float* __restrict__ Bt)
{
    __shared__ float tile[32][33];
    const int bx = blockIdx.x & 7;        // 32-col tile
    const int by = blockIdx.x >> 3;       // 32-row tile
    const int tx = threadIdx.x & 31;
    const int ty = threadIdx.x >> 5;      // 0..7
#pragma unroll
    for (int r = 0; r < 4; ++r)
        tile[ty + 8 * r][tx] = B[(size_t)(by * 32 + ty + 8 * r) * HDIM + bx * 32 + tx];
    __syncthreads();
#pragma unroll
    for (int r = 0; r < 4; ++r)
        Bt[(size_t)(bx * 32 + ty + 8 * r) * HDIM + by * 32 + tx] = tile[tx][ty + 8 * r];
}

// ------------- C = relu(A @ Bt^T + bias)  via V_WMMA_F32_16X16X4_F32 -------
// A:[M,256] row-major, Bt:[256,256] = W^T (so Bt row n holds W[:,n]).
// Each wave computes a 32x64 tile (2x4 WMMA 16x16 tiles). 8 waves / block.
// No divergence anywhere -> EXEC is all ones, as WMMA requires.
__global__ __launch_bounds__(256)
void gemm_relu_wmma_kernel(const float* __restrict__ A, const float* __restrict__ Bt,
                           const float* __restrict__ bias, float* __restrict__ C)
{
    const int lane = threadIdx.x & 31;
    const int wave = threadIdx.x >> 5;
    const int gw   = blockIdx.x * 8 + wave;
    const int tilesN = HDIM / 64;                 // 4
    const int m0 = (gw / tilesN) * 32;
    const int n0 = (gw % tilesN) * 64;
    const int lm = lane & 15;                     // M row (A) / N col (B,C)
    const int lk = (lane >> 4) * 2;               // K sub-offset {0,2}

    v8f acc[2][4];
#pragma unroll
    for (int mi = 0; mi < 2; ++mi)
#pragma unroll
        for (int nt = 0; nt < 4; ++nt)
            acc[mi][nt] = (v8f){0.f, 0.f, 0.f, 0.f, 0.f, 0.f, 0.f, 0.f};

    const float* ap0 = A  + (size_t)(m0 + lm) * HDIM + lk;
    const float* bp0 = Bt + (size_t)(n0 + lm) * HDIM + lk;

    for (int kb = 0; kb < HDIM; kb += 4) {
        v2f a[2];
#pragma unroll
        for (int mi = 0; mi < 2; ++mi)
            a[mi] = *(const v2f*)(ap0 + (size_t)(mi * 16) * HDIM + kb);
        v2f b[4];
#pragma unroll
        for (int nt = 0; nt < 4; ++nt)
            b[nt] = *(const v2f*)(bp0 + (size_t)(nt * 16) * HDIM + kb);
#pragma unroll
        for (int mi = 0; mi < 2; ++mi)
#pragma unroll
            for (int nt = 0; nt < 4; ++nt)
                acc[mi][nt] = __builtin_amdgcn_wmma_f32_16x16x4_f32(
                    false, a[mi], false, b[nt], (short)0, acc[mi][nt], false, false);
    }

    const int row_hi = (lane >> 4) * 8;           // C layout: lanes 16-31 hold M=r+8
#pragma unroll
    for (int mi = 0; mi < 2; ++mi) {
#pragma unroll
        for (int nt = 0; nt < 4; ++nt) {
            const int col = n0 + nt * 16 + lm;
            const float bv = bias[col];
#pragma unroll
            for (int r = 0; r < 8; ++r) {
                const int row = m0 + mi * 16 + row_hi + r;
                const float v = fmaxf(acc[mi][nt][r] + bv, 0.0f);
                C[(size_t)row * HDIM + col] = v;
            }
        }
    }
}

// ------ out = y2 @ W3 + b3 + sum_k w_k * t[idx_k]   (final projection) -----
__global__ __launch_bounds__(256)
void final_kernel(const float* __restrict__ y2, const float* __restrict__ W3,
                  const float* __restrict__ b3, const float* __restrict__ t,
                  const int* __restrict__ idx, const float* __restrict__ w,
                  float* __restrict__ out)
{
    const int gid = blockIdx.x * 256 + threadIdx.x;       // NH*3 threads
    if (gid >= NH * ODIM) return;
    const int q = gid / ODIM;
    const int c = gid % ODIM;
    const float* row = y2 + (size_t)q * HDIM;
    float accum = b3[c];
#pragma unroll 8
    for (int j = 0; j < HDIM; ++j)
        accum = fmaf(row[j], W3[j * ODIM + c], accum);
    const int i0 = idx[q * 3 + 0], i1 = idx[q * 3 + 1], i2 = idx[q * 3 + 2];
    float r = w[q * 3 + 0] * t[i0 * 3 + c];
    r = fmaf(w[q * 3 + 1], t[i1 * 3 + c], r);
    r = fmaf(w[q * 3 + 2], t[i2 * 3 + c], r);
    out[gid] = accum + r;
}

// ---------------------------------------------------------------------------
extern "C" void kernel_launch(void* const* d_in, const int* in_sizes, int n_in,
                              void* d_out, int out_size, void* d_ws, size_t ws_size,
                              hipStream_t stream) {
    const float* emb1   = (const float*)d_in[0];
    const float* l_y1   = (const float*)d_in[1];
    const float* l_pos1 = (const float*)d_in[2];
    const float* h_pos1 = (const float*)d_in[3];
    const float* emb2   = (const float*)d_in[4];
    const float* l_y2   = (const float*)d_in[5];
    const float* l_pos2 = (const float*)d_in[6];
    const float* h_pos2 = (const float*)d_in[7];
    const float* W1     = (const float*)d_in[8];
    const float* b1     = (const float*)d_in[9];
    const float* W2     = (const float*)d_in[10];
    const float* b2     = (const float*)d_in[11];
    const float* W3     = (const float*)d_in[12];
    const float* b3     = (const float*)d_in[13];
    float* out = (float*)d_out;

    // workspace carve-up (~35 MB total)
    char* ws = (char*)d_ws;
    size_t off = 0;
    auto carve = [&](size_t bytes) -> void* {
        void* p = ws + off;
        off += (bytes + 255) & ~(size_t)255;
        return p;
    };
    int*   idx_h = (int*)  carve((size_t)NH * 3 * sizeof(int));
    float* w_h   = (float*)carve((size_t)NH * 3 * sizeof(float));
    int*   idx_l = (int*)  carve((size_t)NL * 3 * sizeof(int));
    float* w_l   = (float*)carve((size_t)NL * 3 * sizeof(float));
    int*   idx_r = (int*)  carve((size_t)NH * 3 * sizeof(int));
    float* w_r   = (float*)carve((size_t)NH * 3 * sizeof(float));
    float* t_l   = (float*)carve((size_t)NL * 3 * sizeof(float));
    float* W1t   = (float*)carve((size_t)HDIM * HDIM * sizeof(float));
    float* W2t   = (float*)carve((size_t)HDIM * HDIM * sizeof(float));
    float* bufA  = (float*)carve((size_t)NH * HDIM * sizeof(float));
    float* bufB  = (float*)carve((size_t)NH * HDIM * sizeof(float));

    // 0) one-time weight transposes (256 KB each; trivial)
    transpose_w_kernel<<<64, 256, 0, stream>>>(W1, W1t);
    transpose_w_kernel<<<64, 256, 0, stream>>>(W2, W2t);

    // 1) the three KNN searches
    knn3_kernel<<<NH / 256, 256, 0, stream>>>(h_pos1, h_pos2, NH, idx_h, w_h);
    knn3_kernel<<<NL / 256, 256, 0, stream>>>(l_pos1, l_pos2, NL, idx_l, w_l);
    knn3_kernel<<<NH / 256, 256, 0, stream>>>(h_pos1, l_pos1, NL, idx_r, w_r);

    // 2) xin = emb1 - interp(emb2)  ;  t = l_y1 - interp(l_y2)
    interp_sub_emb_kernel<<<NH, 256, 0, stream>>>(emb1, emb2, idx_h, w_h, bufA);
    interp_sub3_kernel<<<NL / 256, 256, 0, stream>>>(l_y1, l_y2, idx_l, w_l, t_l, NL);

    // 3) MLP hidden layers on the fp32 WMMA pipe
    const int gemm_blocks = (NH / 32) * (HDIM / 64) / 8;   // 2048 waves / 8 = 256
    gemm_relu_wmma_kernel<<<gemm_blocks, 256, 0, stream>>>(bufA, W1t, b1, bufB);
    gemm_relu_wmma_kernel<<<gemm_blocks, 256, 0, stream>>>(bufB, W2t, b2, bufA);

    // 4) final projection + residual gather + add
    final_kernel<<<(NH * ODIM + 255) / 256, 256, 0, stream>>>(bufA, W3, b3, t_l,
                                                              idx_r, w_r, out);
}